// DetectionLoss_4784593568447
// MI455X (gfx1250) — compile-verified
//
#include <hip/hip_runtime.h>
#include <hip/hip_bf16.h>

// ---------------------------------------------------------------------------
// DetectionLoss for N rows x 2 classes, memory-bound streaming reduction.
//   out = mean_i( lse(o0,o1) - o[label_i] ) + coeff * FN/N
// coeff from confusion counts (TN,FP,FN,TP).
// Design: b128 NT loads (stream > L2), global prefetch, persistent grid,
// per-wave CE reduction via V_WMMA_F32_16X16X4_F32 (B = ones trick),
// deterministic two-pass reduction of block partials.
// ---------------------------------------------------------------------------

typedef float v2f __attribute__((ext_vector_type(2)));
typedef float v4f __attribute__((ext_vector_type(4)));
typedef float v8f __attribute__((ext_vector_type(8)));
typedef int   v2i __attribute__((ext_vector_type(2)));

#define NTHREADS   256
#define NWAVES     (NTHREADS / 32)
#define MAXBLOCKS  2048

// Sum 32 per-lane floats with one WMMA: A(16x4 f32) has the partial in K=0
// (lanes 0-15 -> rows, K slot 0) / K=2 (lanes 16-31) and zeros in K=1,3;
// B(4x16) = all ones => D[m][n] = p[m] + p[m+16] replicated over n.
// Lane 0 holds D[0..7][0] in c[0..7], lane 16 holds D[8..15][0].
// Wave sum = (lane0 c-sum) + (lane16 c-sum).
__device__ inline float wave_sum_wmma(float p) {
  v2f a; a.x = p;    a.y = 0.0f;
  v2f b; b.x = 1.0f; b.y = 1.0f;
  v8f c = {};
  c = __builtin_amdgcn_wmma_f32_16x16x4_f32(
        /*neg_a=*/false, a, /*neg_b=*/false, b,
        /*c_mod=*/(short)0, c, /*reuse_a=*/false, /*reuse_b=*/false);
  float s = ((c[0] + c[1]) + (c[2] + c[3])) + ((c[4] + c[5]) + (c[6] + c[7]));
  s += __shfl_down(s, 16, 32);   // lane0 += lane16
  return s;                      // valid in lane 0 of each wave
}

__global__ void dl_init_counts(unsigned* __restrict__ counts) {
  if (threadIdx.x < 4) counts[threadIdx.x] = 0u;
}

__global__ __launch_bounds__(NTHREADS)
void dl_main(const v4f* __restrict__ outv,   // 2 rows (4 floats) per element
             const v2i* __restrict__ labv,   // 2 labels per element
             int npairs,
             float*    __restrict__ partials,
             unsigned* __restrict__ counts)
{
  const int tid    = blockIdx.x * NTHREADS + threadIdx.x;
  const int stride = gridDim.x * NTHREADS;

  float    ce = 0.0f;
  unsigned tn = 0, fp = 0, fn = 0, tp = 0;

  for (int i = tid; i < npairs; i += stride) {
    // Speculative prefetch a few strides ahead (global_prefetch_b8, NT-ish).
    __builtin_prefetch(&outv[i + 4 * stride], 0, 0);
    __builtin_prefetch(&labv[i + 4 * stride], 0, 0);

    // Stream is read once and exceeds L2 -> non-temporal loads.
    v4f o = __builtin_nontemporal_load(&outv[i]);
    v2i l = __builtin_nontemporal_load(&labv[i]);

    // row 0
    {
      float d  = o.y - o.x;
      bool  pr = d > 0.0f;                 // argmax (first-max tie -> 0)
      bool  lb = l.x != 0;
      float m  = fmaxf(o.x, o.y);
      float ol = lb ? o.y : o.x;
      ce += (m - ol) + __logf(1.0f + __expf(-fabsf(d)));
      tn += (unsigned)(!lb & !pr);
      fp += (unsigned)(!lb &  pr);
      fn += (unsigned)( lb & !pr);
      tp += (unsigned)( lb &  pr);
    }
    // row 1
    {
      float d  = o.w - o.z;
      bool  pr = d > 0.0f;
      bool  lb = l.y != 0;
      float m  = fmaxf(o.z, o.w);
      float ol = lb ? o.w : o.z;
      ce += (m - ol) + __logf(1.0f + __expf(-fabsf(d)));
      tn += (unsigned)(!lb & !pr);
      fp += (unsigned)(!lb &  pr);
      fn += (unsigned)( lb & !pr);
      tp += (unsigned)( lb &  pr);
    }
  }

  // ---- intra-wave CE reduction on the matrix pipe (EXEC all-ones here) ----
  float wsum = wave_sum_wmma(ce);

  __shared__ float    sce[NWAVES];
  __shared__ unsigned scnt[4];
  const int wave = threadIdx.x >> 5;
  const int lane = threadIdx.x & 31;

  if (threadIdx.x < 4) scnt[threadIdx.x] = 0u;
  __syncthreads();

  if (lane == 0) sce[wave] = wsum;
  if (tn) atomicAdd(&scnt[0], tn);
  if (fp) atomicAdd(&scnt[1], fp);
  if (fn) atomicAdd(&scnt[2], fn);
  if (tp) atomicAdd(&scnt[3], tp);
  __syncthreads();

  if (threadIdx.x == 0) {
    float b = 0.0f;
    #pragma unroll
    for (int w = 0; w < NWAVES; ++w) b += sce[w];   // fixed order -> deterministic
    partials[blockIdx.x] = b;
  }
  if (threadIdx.x < 4) atomicAdd(&counts[threadIdx.x], scnt[threadIdx.x]);
}

__global__ __launch_bounds__(NTHREADS)
void dl_finalize(const float* __restrict__ partials, int nparts,
                 const unsigned* __restrict__ counts,
                 float* __restrict__ out, float invN)
{
  __shared__ float s[NTHREADS];
  float acc = 0.0f;
  for (int i = threadIdx.x; i < nparts; i += NTHREADS) acc += partials[i];
  s[threadIdx.x] = acc;
  __syncthreads();
  #pragma unroll
  for (int off = NTHREADS / 2; off > 0; off >>= 1) {
    if (threadIdx.x < off) s[threadIdx.x] += s[threadIdx.x + off];
    __syncthreads();
  }
  if (threadIdx.x == 0) {
    const float TN = (float)counts[0], FP = (float)counts[1];
    const float FN = (float)counts[2], TP = (float)counts[3];
    const bool all_nonzero = (TN != 0.0f) && (FP != 0.0f) && (FN != 0.0f) && (TP != 0.0f);
    const float sens = TP / fmaxf(TP + FN, 1.0f);
    const float prec = TP / fmaxf(TP + FP, 1.0f);
    const float gm   = -0.5f * logf(fmaxf(sens * prec, 1e-30f));
    const float coeff = all_nonzero ? gm : 1.0f;   // LAMBD = 1.0
    out[0] = s[0] * invN + coeff * (FN * invN);
  }
}

extern "C" void kernel_launch(void* const* d_in, const int* in_sizes, int n_in,
                              void* d_out, int out_size, void* d_ws, size_t ws_size,
                              hipStream_t stream) {
  const float* outputs = (const float*)d_in[0];   // [N,2] f32
  const int*   labels  = (const int*)d_in[1];     // [N] i32 (jax default 32-bit)
  float*       out     = (float*)d_out;

  const int N      = in_sizes[1];
  const int npairs = N >> 1;                       // N = 2^24, even

  unsigned* counts   = (unsigned*)d_ws;            // 4 x u32
  float*    partials = (float*)((char*)d_ws + 16); // per-block CE sums

  int nb = MAXBLOCKS;
  if ((size_t)(16 + (size_t)nb * sizeof(float)) > ws_size) {
    nb = (int)((ws_size > 16 ? ws_size - 16 : 4) / sizeof(float));
    if (nb < 1) nb = 1;
  }
  const int maxUseful = (npairs + NTHREADS - 1) / NTHREADS;
  if (nb > maxUseful) nb = maxUseful;
  if (nb < 1) nb = 1;

  dl_init_counts<<<1, 64, 0, stream>>>(counts);
  dl_main<<<nb, NTHREADS, 0, stream>>>((const v4f*)outputs, (const v2i*)labels,
                                       npairs, partials, counts);
  dl_finalize<<<1, NTHREADS, 0, stream>>>(partials, nb, counts, out,
                                          1.0f / (float)N);
}